// GatherGrad_53833120088422
// MI455X (gfx1250) — compile-verified
//
#include <hip/hip_runtime.h>
#include <stdint.h>

#ifndef __has_builtin
#define __has_builtin(x) 0
#endif

#define COLS 1024                 // row length (floats) — matches INPUT_SHAPE[1]
#define WAVES_PER_BLOCK 8         // 256 threads = 8 wave32
#define ROWS_PER_WAVE 8           // each wave scatters 8 index rows
#define CHUNKS_PER_LANE 8         // 1024 floats / (32 lanes * 4 floats)

// Vector-of-4-int type matching the builtin's expected pointee
// ('__attribute__((__vector_size__(4 * sizeof(int)))) int' per the diagnostic).
typedef int v4i_vs __attribute__((vector_size(16)));
typedef __attribute__((address_space(1))) v4i_vs* gptr_v4i;
typedef __attribute__((address_space(3))) v4i_vs* lptr_v4i;

// ---------- CDNA5 async global->LDS DMA, 16 bytes per lane (ASYNCcnt) ----------
// cpol=1 -> TH_LOAD_NT: grad_last is a 512 MB use-once stream; NT keeps it from
// evicting the 200 MB output working set that the L2 atomic units are RMW-ing.
__device__ __forceinline__ void async_copy16(const float* gsrc, float* ldst) {
  // 32-bit LDS offset (low 32 bits of the shared-aperture generic address)
  unsigned lds_off = (unsigned)(uintptr_t)ldst;
#if __has_builtin(__builtin_amdgcn_global_load_async_to_lds_b128)
  __builtin_amdgcn_global_load_async_to_lds_b128(
      (gptr_v4i)(uintptr_t)gsrc,
      (lptr_v4i)lds_off,
      0, /*cpol=TH_LOAD_NT*/ 1);
#else
  asm volatile("global_load_async_to_lds_b128 %0, %1, off th:TH_LOAD_NT"
               :: "v"(lds_off), "v"(gsrc)
               : "memory");
#endif
}

template <int N>
__device__ __forceinline__ void wait_asynccnt_le() {
#if __has_builtin(__builtin_amdgcn_s_wait_asynccnt)
  __builtin_amdgcn_s_wait_asynccnt((short)N);
#else
  asm volatile("s_wait_asynccnt %0" :: "n"(N));
#endif
}

// ---------- no-return f32 atomic add (STOREcnt, device scope) ----------
__device__ __forceinline__ void atom_add_f32(float* p, float v) {
  asm volatile("global_atomic_add_f32 %0, %1, off scope:SCOPE_DEV"
               :: "v"(p), "v"(v)
               : "memory");
}

// ---------- zero the 50000x1024 output (graph-replayed every call) ----------
__global__ __launch_bounds__(256) void GatherGrad_zero_kernel(float* __restrict__ out,
                                                              long n) {
  long n4 = n >> 2;
  float4 z = make_float4(0.f, 0.f, 0.f, 0.f);
  float4* o4 = (float4*)out;
  long stride = (long)gridDim.x * blockDim.x;
  for (long i = (long)blockIdx.x * blockDim.x + threadIdx.x; i < n4; i += stride)
    o4[i] = z;
  if (blockIdx.x == 0 && threadIdx.x == 0)
    for (long t = n4 << 2; t < n; ++t) out[t] = 0.f;
}

// ---------- scatter-add: one wave per index row, double-buffered async DMA ----------
__global__ __launch_bounds__(256) void GatherGrad_scatter_kernel(
    const float* __restrict__ grad, const int* __restrict__ indices,
    float* __restrict__ out, int n_idx) {
  __shared__ float buf[WAVES_PER_BLOCK][2][COLS];   // 64 KB / block (of 320 KB WGP LDS)

  const int wave = threadIdx.x >> 5;
  const int lane = threadIdx.x & 31;
  const int rowBase = (blockIdx.x * WAVES_PER_BLOCK + wave) * ROWS_PER_WAVE;

  // Prefetch row 0 of this wave's strip into buffer 0 (8 async ops = 4 KB).
  if (rowBase < n_idx) {
    const float* src = grad + (size_t)rowBase * COLS;
#pragma unroll
    for (int c = 0; c < CHUNKS_PER_LANE; ++c) {
      int off = (lane + c * 32) * 4;
      async_copy16(src + off, &buf[wave][0][off]);
    }
  }

#pragma unroll
  for (int k = 0; k < ROWS_PER_WAVE; ++k) {
    const int row = rowBase + k;
    if (row >= n_idx) break;

    // Kick off the DMA for the next row into the other buffer, then wait for
    // the current row's 8 async ops (in-order completion => asynccnt<=8 means
    // row k is fully resident in LDS while row k+1 is still in flight).
    if (k + 1 < ROWS_PER_WAVE && row + 1 < n_idx) {
      const float* src = grad + (size_t)(row + 1) * COLS;
#pragma unroll
      for (int c = 0; c < CHUNKS_PER_LANE; ++c) {
        int off = (lane + c * 32) * 4;
        async_copy16(src + off, &buf[wave][(k + 1) & 1][off]);
      }
      wait_asynccnt_le<CHUNKS_PER_LANE>();
    } else {
      wait_asynccnt_le<0>();
    }

    const int d = indices[row];
    float* dst = out + (size_t)d * COLS;
    const float* s = buf[wave][k & 1];
#pragma unroll
    for (int c = 0; c < CHUNKS_PER_LANE; ++c) {
      int off = (lane + c * 32) * 4;
      float4 v = *(const float4*)(s + off);   // ds_load_b128
      atom_add_f32(dst + off + 0, v.x);
      atom_add_f32(dst + off + 1, v.y);
      atom_add_f32(dst + off + 2, v.z);
      atom_add_f32(dst + off + 3, v.w);
    }
  }
}

extern "C" void kernel_launch(void* const* d_in, const int* in_sizes, int n_in,
                              void* d_out, int out_size, void* d_ws, size_t ws_size,
                              hipStream_t stream) {
  const float* grad    = (const float*)d_in[0];   // [n_idx, 1024] f32
  const int*   indices = (const int*)d_in[1];     // [n_idx] int
  float*       out     = (float*)d_out;           // [50000, 1024] f32
  const int n_idx = in_sizes[1];

  // Pass 1: zero the output (poisoned by the harness; accumulates across replays).
  GatherGrad_zero_kernel<<<2048, 256, 0, stream>>>(out, (long)out_size);

  // Pass 2: scatter-add. 64 rows per block (8 waves x 8 rows).
  const int rows_per_block = WAVES_PER_BLOCK * ROWS_PER_WAVE;
  const int blocks = (n_idx + rows_per_block - 1) / rows_per_block;
  if (blocks > 0)
    GatherGrad_scatter_kernel<<<blocks, 256, 0, stream>>>(grad, indices, out, n_idx);
}